// padMask_20495583936830
// MI455X (gfx1250) — compile-verified
//
#include <hip/hip_runtime.h>
#include <stdint.h>

// padMask: out[b,i,j] = (i < len_q[b]) && (j < len_k[b]), as float 1.0/0.0.
// B=32, SEQ=2048 -> 512MB of float output. Pure store-bandwidth problem
// (~22us floor at 23.3 TB/s). All valid rows within a batch are identical,
// so build the row pattern once in LDS and replicate it to HBM with
// gfx1250 async store-from-LDS (GLOBAL_STORE_ASYNC_FROM_LDS_B128, ASYNCcnt).

#define SEQ 2048
#define BATCH 32
#define THREADS 256
#define ROWS_PER_BLOCK 16

typedef float f4  __attribute__((ext_vector_type(4)));
typedef int   v4i __attribute__((ext_vector_type(4)));

typedef __attribute__((address_space(1))) v4i* g_v4i_ptr;  // global int4*
typedef __attribute__((address_space(3))) v4i* l_v4i_ptr;  // LDS int4*

__device__ __forceinline__ void async_store_b128_from_lds(float* gp, const float* lp) {
#if defined(__AMDGCN__) && __has_builtin(__builtin_amdgcn_global_store_async_from_lds_b128)
  // (global dst as AS1 int4*, lds src as AS3 int4*, imm offset, imm cpol)
  __builtin_amdgcn_global_store_async_from_lds_b128(
      (g_v4i_ptr)(v4i*)gp,
      (l_v4i_ptr)(v4i*)lp,
      0, 0);
#else
  // ISA operand order: VADDR (64-bit global addr), VSRC (LDS byte addr), GV mode.
  asm volatile("global_store_async_from_lds_b128 %0, %1, off"
               :
               : "v"(gp), "v"((uint32_t)(uint64_t)lp)
               : "memory");
#endif
}

__device__ __forceinline__ void wait_asynccnt0() {
#if defined(__AMDGCN__) && __has_builtin(__builtin_amdgcn_s_wait_asynccnt)
  __builtin_amdgcn_s_wait_asynccnt(0);
#else
  asm volatile("s_wait_asynccnt 0x0" ::: "memory");
#endif
}

__global__ __launch_bounds__(THREADS) void padmask_async_kernel(
    const int* __restrict__ len_q,
    const int* __restrict__ len_k,
    float* __restrict__ out) {
  // LDS: [0, SEQ) = "ones" row pattern for this batch, [SEQ, 2*SEQ) = zero row.
  __shared__ __align__(16) float lds_rows[2 * SEQ];

  const int b   = blockIdx.y;      // batch (uniform per block -> s_load of lens)
  const int lq  = len_q[b];
  const int lk  = len_k[b];
  const int tid = threadIdx.x;

  // Build both row patterns: 512 float4 chunks per row, 2 chunks per thread.
#pragma unroll
  for (int s = 0; s < 2; ++s) {
    const int j0 = (tid + s * THREADS) * 4;  // first column of this chunk
    f4 ones;
    ones.x = (j0 + 0) < lk ? 1.0f : 0.0f;
    ones.y = (j0 + 1) < lk ? 1.0f : 0.0f;
    ones.z = (j0 + 2) < lk ? 1.0f : 0.0f;
    ones.w = (j0 + 3) < lk ? 1.0f : 0.0f;
    *(f4*)&lds_rows[j0] = ones;
    f4 z = {0.0f, 0.0f, 0.0f, 0.0f};
    *(f4*)&lds_rows[SEQ + j0] = z;
  }
  __syncthreads();  // LDS pattern visible before async engine reads it

  const int i0 = blockIdx.x * ROWS_PER_BLOCK;
#pragma unroll 4
  for (int r = 0; r < ROWS_PER_BLOCK; ++r) {
    const int i = i0 + r;
    const float* src = (i < lq) ? &lds_rows[0] : &lds_rows[SEQ];
    float* dst = out + ((size_t)b * SEQ + (size_t)i) * (size_t)SEQ;
#pragma unroll
    for (int s = 0; s < 2; ++s) {
      const int off = (tid + s * THREADS) * 4;
      async_store_b128_from_lds(dst + off, src + off);
    }
  }

  // S_ENDPGM implicitly waits all counters, but drain explicitly while LDS
  // contents are provably untouched (nothing writes LDS after the barrier).
  wait_asynccnt0();
}

extern "C" void kernel_launch(void* const* d_in, const int* in_sizes, int n_in,
                              void* d_out, int out_size, void* d_ws, size_t ws_size,
                              hipStream_t stream) {
  const int* len_q = (const int*)d_in[0];
  const int* len_k = (const int*)d_in[1];
  float* out = (float*)d_out;

  dim3 grid(SEQ / ROWS_PER_BLOCK, BATCH);  // 128 x 32 = 4096 blocks
  padmask_async_kernel<<<grid, THREADS, 0, stream>>>(len_q, len_k, out);
}